// attention_19645180412780
// MI455X (gfx1250) — compile-verified
//
#include <hip/hip_runtime.h>

typedef __attribute__((ext_vector_type(2))) float v2f;
typedef __attribute__((ext_vector_type(8))) float v8f;

#define T_DIM 2048
#define D_DIM 128
#define CHUNK 64   // timesteps per streamed chunk (64*128*4 = 32KB)

// One workgroup per batch. 256 threads = 8 wave32s.
// Single streaming pass over x (HBM-bound: ~256MB once ~= 11.5us floor):
//   phase 1 (per chunk): logits e_t = tanh(x_t . W + b_t), p_t = exp(e_t)
//           via float4 coalesced loads + wave shuffle reduction; p -> LDS.
//   phase 2 (per chunk): acc[d] += p_t * x[t,d] as V_WMMA_F32_16X16X4_F32,
//           B = natural row-major 4x16 tile of x (re-read, WGP$/L2 hot),
//           A = p broadcast into all 16 rows, C accumulates across all T.
// Final: out = acc / sum(p). tanh bounds logits so softmax needs no max pass.
__global__ __launch_bounds__(256) void attn_pool_kernel(
    const float* __restrict__ x,
    const float* __restrict__ W,
    const float* __restrict__ bias,
    float* __restrict__ out)
{
    __shared__ float pbuf[2][CHUNK];
    __shared__ float s_shared;

    const int b    = blockIdx.x;
    const int tid  = threadIdx.x;
    const int wave = tid >> 5;        // 0..7
    const int lane = tid & 31;        // 0..31
    const int half = lane >> 4;       // 0 or 1 (WMMA K0/K1 vs K2/K3 half-wave)
    const int n    = lane & 15;       // WMMA N index / output column in tile
    const int d0   = wave * 16;       // each wave owns one 16-wide d tile

    const float* xb = x + (size_t)b * T_DIM * D_DIM;

    if (tid == 0) s_shared = 0.0f;

    // Each lane holds its 4 W values for the dot-product phase.
    const float4 w4 = reinterpret_cast<const float4*>(W)[lane];

    v8f   c = {};         // WMMA accumulator (all 16 rows identical)
    float s_local = 0.0f; // per-wave sum of p (lane-uniform)

    __syncthreads();

    for (int base = 0; base < T_DIM; base += CHUNK) {
        const int cur = (base / CHUNK) & 1;

        // Prefetch next chunk: 256 threads x 128B = 32KB (gfx1250 global_prefetch_b8)
        if (base + CHUNK < T_DIM) {
            __builtin_prefetch(xb + (size_t)(base + CHUNK) * D_DIM + tid * 32, 0, 0);
        }

        // ---- Phase 1: one wave per timestep, 8 sub-iterations -> 64 logits ----
        for (int j = 0; j < 8; ++j) {
            const int t = base + j * 8 + wave;
            const float4 xv =
                reinterpret_cast<const float4*>(xb + (size_t)t * D_DIM)[lane];
            float partial = xv.x * w4.x + xv.y * w4.y + xv.z * w4.z + xv.w * w4.w;
            // wave32 all-reduce
            partial += __shfl_xor(partial, 1, 32);
            partial += __shfl_xor(partial, 2, 32);
            partial += __shfl_xor(partial, 4, 32);
            partial += __shfl_xor(partial, 8, 32);
            partial += __shfl_xor(partial, 16, 32);
            const float e = tanhf(partial + bias[t]);
            const float p = __expf(e);   // e in (-1,1): no max-subtraction needed
            s_local += p;
            if (lane == 0) pbuf[cur][j * 8 + wave] = p;
        }
        __syncthreads();  // p visible; double buffering makes this the only barrier

        // ---- Phase 2: WMMA weighted accumulation over the 64 timesteps ----
        for (int i = 0; i < 16; ++i) {
            const int t0 = base + 4 * i;
            const float p0 = pbuf[cur][4 * i + 0];
            const float p1 = pbuf[cur][4 * i + 1];
            const float p2 = pbuf[cur][4 * i + 2];
            const float p3 = pbuf[cur][4 * i + 3];

            // A (16x4, rows identical): VGPR0 = K0|K2, VGPR1 = K1|K3 split at lane 16
            v2f a;
            a.x = half ? p2 : p0;
            a.y = half ? p3 : p1;

            // B (4x16): VGPR0 lanes0-15 = row t0, lanes16-31 = row t0+2; VGPR1 = +1
            const float* col = xb + (size_t)(t0 + 2 * half) * D_DIM + d0 + n;
            v2f bb;
            bb.x = col[0];
            bb.y = col[D_DIM];

            c = __builtin_amdgcn_wmma_f32_16x16x4_f32(
                false, a, false, bb, (short)0, c, false, false);
        }
    }

    if (lane == 0) atomicAdd(&s_shared, s_local);
    __syncthreads();

    const float inv = 1.0f / s_shared;
    if (lane < 16) {
        // C layout: VGPR0, lanes 0-15 = (M=0, N=lane); all rows identical
        out[(size_t)b * D_DIM + d0 + lane] = c[0] * inv;
    }
}

extern "C" void kernel_launch(void* const* d_in, const int* in_sizes, int n_in,
                              void* d_out, int out_size, void* d_ws, size_t ws_size,
                              hipStream_t stream) {
    const float* x    = (const float*)d_in[0];  // [B, T, D] f32
    const float* W    = (const float*)d_in[1];  // [D, 1]  f32
    const float* bias = (const float*)d_in[2];  // [T, 1]  f32
    float* out = (float*)d_out;                 // [B, D]  f32

    const int B = in_sizes[0] / (T_DIM * D_DIM);
    attn_pool_kernel<<<B, 256, 0, stream>>>(x, W, bias, out);
}